// HashNGramEmbedding_9998683865342
// MI455X (gfx1250) — compile-verified
//
#include <hip/hip_runtime.h>
#include <stdint.h>

// HashNGramEmbedding for MI455X (gfx1250).
//
// Roofline: ~402 MB of random 1KB embedding-row reads + 67 MB of writes vs
// ~0.1 GFLOP of adds -> pure HBM gather bandwidth (~20us floor at 23.3 TB/s).
// The CDNA5 Tensor Data Mover *gather mode* (D#.gather_mode=1, 32-bit row
// indices, 8 rows/descriptor) is the hardware path built for this: the DMA
// engine gathers rows into LDS (TENSORcnt-tracked), waves only do the 6-way
// sum and a coalesced store.
//
// Per workgroup (8 positions): waves 0..5 each own one table - compute the
// rolling hashes, build one gather descriptor, issue one TENSOR_LOAD_TO_LDS,
// wait on their own TENSORcnt. Barrier. Then wave w sums position w's six
// staged rows from LDS and stores a contiguous 1KB output row.

#define HASH_BASE 257
#define VOCAB     100000
#define EMBED     256
#define SEQ_LEN   8192
#define NTAB      6
#define NPOS      8        // positions per workgroup (= rows per TDM gather)
#define THREADS   256      // 8 wave32

typedef __attribute__((ext_vector_type(4))) unsigned int u32x4;
typedef __attribute__((ext_vector_type(8))) int          i32x8;
typedef __attribute__((ext_vector_type(4))) int          i32x4;

__global__ __launch_bounds__(THREADS)
void hash_ngram_embed_kernel(const int*   __restrict__ tokens,
                             const float* __restrict__ E3,
                             const float* __restrict__ E4,
                             const float* __restrict__ E5,
                             const float* __restrict__ E6,
                             const float* __restrict__ E7,
                             const float* __restrict__ E8,
                             float*       __restrict__ out)
{
    // 6 tables * 8 positions * 256 floats = 48 KB staged by the TDM.
    __shared__ __align__(16) float smem[NTAB * NPOS * EMBED];

    const int tid     = (int)threadIdx.x;
    const int wgStart = (int)blockIdx.x * NPOS;

    // ---- Waves 0..5: one TDM gather descriptor per wave (one per table).
    if (tid < NTAB * 32) {
        const int t = __builtin_amdgcn_readfirstlane(tid >> 5); // uniform table id 0..5
        const int n = t + 3;                                    // ngram size

        const int l   = tid & (NPOS - 1);   // lanes 0..7 carry real positions
        const int p   = wgStart + l;
        const int s   = p & (SEQ_LEN - 1);  // position within the sequence row
        const int raw = tokens[p];

        // Rolling Horner hash, matching the reference exactly:
        //   h = b[s]; for j=1..n-1: h = (h*257 + b[s-j]) % VOCAB
        // Branch-free: clamp the address into the row (always valid) and
        // select 0 for the padded positions, so all loads issue back-to-back.
        int h    = raw;
        int hsel = raw;
        #pragma unroll
        for (int j = 1; j <= 7; ++j) {
            const int cj = (s < j) ? s : j;        // clamp to row start
            const int v  = tokens[p - cj];         // unconditional, in-bounds
            const int bj = (s >= j) ? v : 0;       // zero-pad select
            h = (h * HASH_BASE + bj) % VOCAB;
            if (j == n - 1) hsel = h;              // pick this wave's ngram hash
        }
        const int id = (s < n - 1) ? raw : hsel;   // raw byte for warm-up prefix

        // Uniform scalar select of this wave's table base (stays in SGPRs).
        const float* tab = E3;
        if (t == 1) tab = E4;
        if (t == 2) tab = E5;
        if (t == 3) tab = E6;
        if (t == 4) tab = E7;
        if (t == 5) tab = E8;

        // Low 32 bits of the generic LDS pointer == LDS byte offset
        // (the shared aperture lives entirely in the high dword on CDNA5).
        const uint32_t lds  = (uint32_t)(uintptr_t)(&smem[0])
                            + (uint32_t)t * (NPOS * EMBED * 4);
        const uint64_t ga   = (uint64_t)(uintptr_t)tab;

        // ---- D# group 0 (ISA 8.3): count=1, gather_mode=1, 32b indices,
        //      lds_addr, global_addr[56:0], type=2.
        u32x4 g0;
        g0[0] = 0xC0000001u;                                 // gather|idx32|count=1
        g0[1] = lds;                                         // lds_addr
        g0[2] = (uint32_t)ga;                                // global_addr[31:0]
        g0[3] = (uint32_t)((ga >> 32) & 0x01FFFFFFu)         // global_addr[56:32]
                | 0x80000000u;                               // type=2 ("image")

        // ---- D# group 1 (ISA 8.4): data_size=4B, tensor_dim0=EMBED,
        //      tensor_dim1=VOCAB, tile_dim0=EMBED, tile_dim1=#indices,
        //      tensor_dim0_stride=EMBED. No multicast, no padding.
        i32x8 g1;
        g1[0] = 2 << 16;                          // data_size=2 -> 4 bytes
        g1[1] = (EMBED & 0xFFFF) << 16;           // tensor_dim0[15:0]
        g1[2] = (VOCAB & 0xFFFF) << 16;           // dim0[31:16]=0 | tensor_dim1[15:0]
        g1[3] = (VOCAB >> 16) | (EMBED << 16);    // tensor_dim1[31:16] | tile_dim0
        g1[4] = NPOS;                             // tile_dim1 = #valid indices
        g1[5] = EMBED;                            // tensor_dim0_stride[31:0]
        g1[6] = 0;
        g1[7] = 0;

        // ---- D# groups 2/3 (gather mode, 32-bit): row_index_0..7 = hash ids
        //      of the 8 positions, pulled to SGPRs via readlane.
        i32x4 g2, g3;
        g2[0] = __builtin_amdgcn_readlane(id, 0);
        g2[1] = __builtin_amdgcn_readlane(id, 1);
        g2[2] = __builtin_amdgcn_readlane(id, 2);
        g2[3] = __builtin_amdgcn_readlane(id, 3);
        g3[0] = __builtin_amdgcn_readlane(id, 4);
        g3[1] = __builtin_amdgcn_readlane(id, 5);
        g3[2] = __builtin_amdgcn_readlane(id, 6);
        g3[3] = __builtin_amdgcn_readlane(id, 7);

        // 6-arg toolchain form: the 5th operand is the unused VADDR4 slot.
        i32x8 g4 = { 0, 0, 0, 0, 0, 0, 0, 0 };

        __builtin_amdgcn_tensor_load_to_lds(g0, g1, g2, g3, g4, 0);

        // Each wave waits only on its own single descriptor.
        __builtin_amdgcn_s_wait_tensorcnt(0);
    }
    __syncthreads();

    // ---- Sum phase: wave w owns position wgStart+w; lane handles 8 floats.
    const int posL = tid >> 5;       // 0..7
    const int lane = tid & 31;       // 0..31, 8 floats each -> 256

    float a0 = 0.f, a1 = 0.f, a2 = 0.f, a3 = 0.f;
    float a4 = 0.f, a5 = 0.f, a6 = 0.f, a7 = 0.f;
    #pragma unroll
    for (int t = 0; t < NTAB; ++t) {
        const float4* row =
            (const float4*)&smem[(t * NPOS + posL) * EMBED];   // 1KB row in LDS
        const float4 v0 = row[lane * 2 + 0];                    // ds_load_b128
        const float4 v1 = row[lane * 2 + 1];                    // ds_load_b128
        a0 += v0.x; a1 += v0.y; a2 += v0.z; a3 += v0.w;
        a4 += v1.x; a5 += v1.y; a6 += v1.z; a7 += v1.w;
    }
    const float sc = 1.0f / (float)NTAB;
    const float4 o0 = make_float4(a0 * sc, a1 * sc, a2 * sc, a3 * sc);
    const float4 o1 = make_float4(a4 * sc, a5 * sc, a6 * sc, a7 * sc);

    // Wave writes a contiguous 1KB row: fully coalesced global_store_b128.
    float4* op = (float4*)(out + (size_t)(wgStart + posL) * EMBED + lane * 8);
    op[0] = o0;
    op[1] = o1;
}

extern "C" void kernel_launch(void* const* d_in, const int* in_sizes, int n_in,
                              void* d_out, int out_size, void* d_ws, size_t ws_size,
                              hipStream_t stream)
{
    const int*   tokens = (const int*)d_in[0];
    const float* E3     = (const float*)d_in[1];
    const float* E4     = (const float*)d_in[2];
    const float* E5     = (const float*)d_in[3];
    const float* E6     = (const float*)d_in[4];
    const float* E7     = (const float*)d_in[5];
    const float* E8     = (const float*)d_in[6];
    float*       out    = (float*)d_out;

    const int totalPos = in_sizes[0];           // B*S = 65536
    const int blocks   = totalPos / NPOS;       // 8192 workgroups

    hash_ngram_embed_kernel<<<dim3(blocks), dim3(THREADS), 0, stream>>>(
        tokens, E3, E4, E5, E6, E7, E8, out);
}